// Attention_61392262529362
// MI455X (gfx1250) — compile-verified
//
#include <hip/hip_runtime.h>

typedef __attribute__((ext_vector_type(2))) float v2f;
typedef __attribute__((ext_vector_type(8))) float v8f;

constexpr int B  = 4;
constexpr int LQ = 256;
constexpr int LK = 1024;
constexpr int QD = 512;   // == KD == VD
constexpr int VD = 512;
constexpr int H  = 128;

// Native CDNA5 transcendental tanh (trans pipe, co-executes with VALU)
__device__ __forceinline__ float fast_tanh(float x) {
    float y;
    asm("v_tanh_f32 %0, %1" : "=v"(y) : "v"(x));
    return y;
}

// ---------------------------------------------------------------------------
// Y[M,H] = X[M,D] * W^T, W stored (H,D) row-major. D=512, H=128, f32 WMMA.
// blockDim = 256 (8 waves). Wave w computes tile [blockIdx.x*16, w*16].
// ---------------------------------------------------------------------------
__global__ void __launch_bounds__(256)
proj_kernel(const float* __restrict__ X, const float* __restrict__ W,
            float* __restrict__ Y) {
    const int wv   = threadIdx.x >> 5;
    const int lane = threadIdx.x & 31;
    const int m0   = blockIdx.x * 16;
    const int n0   = wv * 16;
    const int r    = lane & 15;
    const int kk   = (lane >> 4) << 1;     // 0 or 2

    const float* Xr = X + (size_t)(m0 + r) * QD + kk;   // A row, K-offset
    const float* Wr = W + (size_t)(n0 + r) * QD + kk;   // B[k][n]=W[n][k]

    v8f acc = {0.f, 0.f, 0.f, 0.f, 0.f, 0.f, 0.f, 0.f};
    for (int k0 = 0; k0 < QD; k0 += 4) {
        v2f a = *(const v2f*)(Xr + k0);
        v2f b = *(const v2f*)(Wr + k0);
        acc = __builtin_amdgcn_wmma_f32_16x16x4_f32(
            false, a, false, b, (short)0, acc, false, false);
    }
    const int rbase = m0 + ((lane >> 4) << 3);   // +8 for upper half-wave
    const int c     = n0 + r;
#pragma unroll
    for (int i = 0; i < 8; ++i)
        Y[(size_t)(rbase + i) * H + c] = acc[i];
}

// ---------------------------------------------------------------------------
// Per (b,q): scores over k=0..LK-1 via v^T tanh(q_h + k_h), then softmax.
// blockDim=256, each thread owns 4 k-values. Writes weights row to d_out.
// ---------------------------------------------------------------------------
__global__ void __launch_bounds__(256)
score_softmax_kernel(const float* __restrict__ q_h,
                     const float* __restrict__ k_h,
                     const float* __restrict__ v,
                     float* __restrict__ wout) {
    __shared__ __align__(16) float sh_qh[H];
    __shared__ __align__(16) float sh_v[H];
    __shared__ float red[256];

    const int tid = threadIdx.x;
    const int bq  = blockIdx.x;            // 0 .. B*LQ-1
    const int b   = bq / LQ;

    if (tid < H) {
        sh_qh[tid] = q_h[(size_t)bq * H + tid];
        sh_v[tid]  = v[tid];
    }
    __syncthreads();

    float s[4];
#pragma unroll
    for (int i = 0; i < 4; ++i) {
        const int k = i * 256 + tid;
        const float4* kr = (const float4*)(k_h + ((size_t)b * LK + k) * H);
        float acc = 0.f;
#pragma unroll 4
        for (int h = 0; h < H / 4; ++h) {
            float4 kv = kr[h];
            float4 qv = ((const float4*)sh_qh)[h];
            acc = fmaf(sh_v[4 * h + 0], fast_tanh(qv.x + kv.x), acc);
            acc = fmaf(sh_v[4 * h + 1], fast_tanh(qv.y + kv.y), acc);
            acc = fmaf(sh_v[4 * h + 2], fast_tanh(qv.z + kv.z), acc);
            acc = fmaf(sh_v[4 * h + 3], fast_tanh(qv.w + kv.w), acc);
        }
        s[i] = acc;
    }

    // block-wide max
    float m = fmaxf(fmaxf(s[0], s[1]), fmaxf(s[2], s[3]));
    red[tid] = m;
    __syncthreads();
    for (int off = 128; off > 0; off >>= 1) {
        if (tid < off) red[tid] = fmaxf(red[tid], red[tid + off]);
        __syncthreads();
    }
    const float rowmax = red[0];
    __syncthreads();

    // exp + block-wide sum
    float e[4], p = 0.f;
#pragma unroll
    for (int i = 0; i < 4; ++i) { e[i] = __expf(s[i] - rowmax); p += e[i]; }
    red[tid] = p;
    __syncthreads();
    for (int off = 128; off > 0; off >>= 1) {
        if (tid < off) red[tid] += red[tid + off];
        __syncthreads();
    }
    const float inv = 1.0f / red[0];

#pragma unroll
    for (int i = 0; i < 4; ++i)
        wout[(size_t)bq * LK + i * 256 + tid] = e[i] * inv;
}

// ---------------------------------------------------------------------------
// out[b, LQ, VD] = weights[b, LQ, LK] * values[b, LK, VD]  (f32 WMMA)
// grid = (B*LQ/16, VD/128), blockDim=256 (8 waves, one 16x16 tile each)
// ---------------------------------------------------------------------------
__global__ void __launch_bounds__(256)
out_gemm_kernel(const float* __restrict__ weights,
                const float* __restrict__ values,
                float* __restrict__ out) {
    const int wv   = threadIdx.x >> 5;
    const int lane = threadIdx.x & 31;
    const int b    = blockIdx.x / (LQ / 16);
    const int m0   = (blockIdx.x % (LQ / 16)) * 16;
    const int n0   = blockIdx.y * 128 + wv * 16;
    const int r    = lane & 15;
    const int kk   = (lane >> 4) << 1;

    const float* Ar = weights + ((size_t)b * LQ + m0 + r) * LK + kk;
    const float* Vb = values + (size_t)b * LK * VD + n0 + r;

    v8f acc = {0.f, 0.f, 0.f, 0.f, 0.f, 0.f, 0.f, 0.f};
    for (int k0 = 0; k0 < LK; k0 += 4) {
        v2f a = *(const v2f*)(Ar + k0);
        v2f bf;
        bf.x = Vb[(size_t)(k0 + kk + 0) * VD];
        bf.y = Vb[(size_t)(k0 + kk + 1) * VD];
        acc = __builtin_amdgcn_wmma_f32_16x16x4_f32(
            false, a, false, bf, (short)0, acc, false, false);
    }
    const int rbase = m0 + ((lane >> 4) << 3);
    float* Yo = out + (size_t)b * LQ * VD + n0 + r;
#pragma unroll
    for (int i = 0; i < 8; ++i)
        Yo[(size_t)(rbase + i) * VD] = acc[i];
}

// ---------------------------------------------------------------------------
extern "C" void kernel_launch(void* const* d_in, const int* in_sizes, int n_in,
                              void* d_out, int out_size, void* d_ws, size_t ws_size,
                              hipStream_t stream) {
    const float* query  = (const float*)d_in[0];
    const float* keys   = (const float*)d_in[1];
    const float* values = (const float*)d_in[2];
    const float* Wq     = (const float*)d_in[3];
    const float* Wk     = (const float*)d_in[4];
    const float* v      = (const float*)d_in[5];

    float* out     = (float*)d_out;                       // [B,LQ,VD]
    float* weights = out + (size_t)B * LQ * VD;           // [B,LQ,LK]

    float* q_h = (float*)d_ws;                            // [B*LQ, H] 512KB
    float* k_h = q_h + (size_t)B * LQ * H;                // [B*LK, H] 2MB

    proj_kernel<<<dim3(B * LQ / 16), 256, 0, stream>>>(query, Wq, q_h);
    proj_kernel<<<dim3(B * LK / 16), 256, 0, stream>>>(keys, Wk, k_h);
    score_softmax_kernel<<<dim3(B * LQ), 256, 0, stream>>>(q_h, k_h, v, weights);
    out_gemm_kernel<<<dim3(B * (LQ / 16), VD / 128), 256, 0, stream>>>(
        weights, values, out);
}